// HeteroRGCN_62801011802252
// MI455X (gfx1250) — compile-verified
//
#include <hip/hip_runtime.h>
#include <math.h>

typedef __attribute__((ext_vector_type(2))) float v2f;
typedef __attribute__((ext_vector_type(8))) float v8f;

// ---------------------------------------------------------------------------
// Zero workspace (agg1, deg, h, hW2, agg2) -- deterministic per call.
// ---------------------------------------------------------------------------
__global__ __launch_bounds__(1024) void zero_ws_kernel(float* __restrict__ ws, long long n) {
    long long i = (long long)blockIdx.x * blockDim.x + threadIdx.x;
    if (i < n) ws[i] = 0.0f;
}

// ---------------------------------------------------------------------------
// Layer-1 edge kernel: m = x[src] @ W1[etype]  (7x16), scatter-add into agg1,
// count degree. 16 lanes per edge (one per output channel); W1 cached in LDS.
// ---------------------------------------------------------------------------
#define L1_ITER 8
__global__ __launch_bounds__(256) void edge_l1_kernel(
    const float* __restrict__ x, const int* __restrict__ src,
    const int* __restrict__ dst, const int* __restrict__ et,
    const float* __restrict__ W1, float* __restrict__ agg1,
    float* __restrict__ deg, int nEdges)
{
    __shared__ float w1s[16 * 7 * 16];
    for (int i = threadIdx.x; i < 16 * 7 * 16; i += 256) w1s[i] = W1[i];
    __syncthreads();

    long long base = (long long)blockIdx.x * (256LL * L1_ITER);
    #pragma unroll
    for (int it = 0; it < L1_ITER; ++it) {
        long long task = base + (long long)it * 256 + threadIdx.x;
        int e = (int)(task >> 4);      // edge index
        int o = (int)(task & 15);      // output channel
        if (e < nEdges) {
            int s = src[e], d = dst[e], r = et[e];
            const float* xr = x + (long long)s * 7;
            const float* wr = w1s + r * (7 * 16) + o;
            float m = xr[0] * wr[0]   + xr[1] * wr[16] + xr[2] * wr[32]
                    + xr[3] * wr[48]  + xr[4] * wr[64] + xr[5] * wr[80]
                    + xr[6] * wr[96];
            unsafeAtomicAdd(&agg1[(long long)d * 16 + o], m);
            if (o == 0) unsafeAtomicAdd(&deg[d], 1.0f);
        }
    }
}

// ---------------------------------------------------------------------------
// Convert deg -> 1/max(deg,1) in place (full IEEE divide, done once per node
// so the hot WMMA/node kernels only multiply).
// ---------------------------------------------------------------------------
__global__ __launch_bounds__(256) void rdeg_kernel(float* __restrict__ deg, int nNodes) {
    int n = blockIdx.x * blockDim.x + threadIdx.x;
    if (n < nNodes) deg[n] = 1.0f / fmaxf(deg[n], 1.0f);
}

// ---------------------------------------------------------------------------
// Layer-1 node kernel (WMMA): h = relu(agg1*rdeg + x @ root1 + b1)
// One wave per 16-node tile. A = x tile (16x7, K padded to 8 over two
// 16x16x4 WMMA steps), B = root1 (7x16), C preloaded with mean-agg + bias.
// ---------------------------------------------------------------------------
__global__ __launch_bounds__(256) void node_l1_kernel(
    const float* __restrict__ x, const float* __restrict__ root1,
    const float* __restrict__ bias1, const float* __restrict__ agg1,
    const float* __restrict__ rdeg, float* __restrict__ h, int nTiles)
{
    int lane = threadIdx.x & 31;
    int wave = threadIdx.x >> 5;
    int tile = blockIdx.x * 8 + wave;
    if (tile >= nTiles) return;
    int n0   = tile * 16;
    int half = lane >> 4;     // K-pair selector for A/B, row-half for C/D
    int col  = lane & 15;     // A row (M) / B,C,D column (N)

    // A: 32-bit A-matrix 16x4 layout: lanes 0-15 => K=0,1 ; lanes 16-31 => K=2,3
    const float* xr = x + (long long)(n0 + col) * 7;
    v2f a0, a1;
    a0.x = xr[2 * half];
    a0.y = xr[2 * half + 1];
    a1.x = xr[4 + 2 * half];            // K=4 (half 0) / K=6 (half 1)
    a1.y = half ? 0.0f : xr[5];         // K=5 / K=7 (zero pad)

    // B: 4x16 layout mirrors A: lanes 0-15 => rows K=0,1 ; lanes 16-31 => K=2,3
    v2f b0, b1;
    b0.x = root1[(2 * half) * 16 + col];
    b0.y = root1[(2 * half + 1) * 16 + col];
    b1.x = root1[(4 + 2 * half) * 16 + col];
    b1.y = half ? 0.0f : root1[5 * 16 + col];

    // C: VGPR v, lanes 0-15 => M=v ; lanes 16-31 => M=8+v
    v8f c;
    #pragma unroll
    for (int v = 0; v < 8; ++v) {
        int nd = n0 + half * 8 + v;
        c[v] = agg1[(long long)nd * 16 + col] * rdeg[nd] + bias1[col];
    }

    v8f dacc = __builtin_amdgcn_wmma_f32_16x16x4_f32(false, a0, false, b0,
                                                     (short)0, c, false, false);
    dacc     = __builtin_amdgcn_wmma_f32_16x16x4_f32(false, a1, false, b1,
                                                     (short)0, dacc, false, false);

    #pragma unroll
    for (int v = 0; v < 8; ++v) {
        int nd = n0 + half * 8 + v;
        h[(long long)nd * 16 + col] = fmaxf(dacc[v], 0.0f);
    }
}

// ---------------------------------------------------------------------------
// Layer-2 message precompute (WMMA): hW2[n, r*2+o] = h[n,:] @ W2[r,:,o].
// Output is [N,32]; K=16 via 4 chained 16x16x4 WMMA steps, 2 column blocks.
// ---------------------------------------------------------------------------
__global__ __launch_bounds__(256) void hw2_kernel(
    const float* __restrict__ h, const float* __restrict__ W2,
    float* __restrict__ hW2, int nTiles)
{
    int lane = threadIdx.x & 31;
    int wave = threadIdx.x >> 5;
    int tile = blockIdx.x * 8 + wave;
    if (tile >= nTiles) return;
    int n0   = tile * 16;
    int half = lane >> 4;
    int col  = lane & 15;

    const float* hr = h + (long long)(n0 + col) * 16;
    v2f a[4];
    #pragma unroll
    for (int s = 0; s < 4; ++s) {
        a[s].x = hr[4 * s + 2 * half];
        a[s].y = hr[4 * s + 2 * half + 1];
    }

    #pragma unroll
    for (int cb = 0; cb < 2; ++cb) {
        int cc = cb * 16 + col;        // flat output column in [0,32)
        int r  = cc >> 1;              // relation
        int o  = cc & 1;               // output feature
        v8f dacc = {};
        #pragma unroll
        for (int s = 0; s < 4; ++s) {
            int k0 = 4 * s + 2 * half;
            v2f b;
            b.x = W2[(r * 16 + k0) * 2 + o];
            b.y = W2[(r * 16 + k0 + 1) * 2 + o];
            dacc = __builtin_amdgcn_wmma_f32_16x16x4_f32(false, a[s], false, b,
                                                         (short)0, dacc, false, false);
        }
        #pragma unroll
        for (int v = 0; v < 8; ++v) {
            int nd = n0 + half * 8 + v;
            hW2[(long long)nd * 32 + cc] = dacc[v];
        }
    }
}

// ---------------------------------------------------------------------------
// Layer-2 edge kernel: gather 8B from L2-resident hW2, scatter-add 2 floats.
// ---------------------------------------------------------------------------
__global__ __launch_bounds__(256) void edge_l2_kernel(
    const int* __restrict__ src, const int* __restrict__ dst,
    const int* __restrict__ et, const float* __restrict__ hW2,
    float* __restrict__ agg2, int nEdges)
{
    int e = blockIdx.x * blockDim.x + threadIdx.x;
    if (e >= nEdges) return;
    int s = src[e], d = dst[e], r = et[e];
    const float* p = hW2 + (long long)s * 32 + r * 2;
    unsafeAtomicAdd(&agg2[(long long)d * 2 + 0], p[0]);
    unsafeAtomicAdd(&agg2[(long long)d * 2 + 1], p[1]);
}

// ---------------------------------------------------------------------------
// Final: out = log_softmax(agg2*rdeg + h @ root2 + b2)
// ---------------------------------------------------------------------------
__global__ __launch_bounds__(256) void final_kernel(
    const float* __restrict__ h, const float* __restrict__ root2,
    const float* __restrict__ b2, const float* __restrict__ agg2,
    const float* __restrict__ rdeg, float* __restrict__ out, int nNodes)
{
    int n = blockIdx.x * blockDim.x + threadIdx.x;
    if (n >= nNodes) return;
    float rd = rdeg[n];
    float o0 = agg2[(long long)n * 2 + 0] * rd + b2[0];
    float o1 = agg2[(long long)n * 2 + 1] * rd + b2[1];
    const float* hr = h + (long long)n * 16;
    #pragma unroll
    for (int k = 0; k < 16; ++k) {
        float hv = hr[k];
        o0 += hv * root2[k * 2 + 0];
        o1 += hv * root2[k * 2 + 1];
    }
    float mx = fmaxf(o0, o1);
    float ls = mx + logf(expf(o0 - mx) + expf(o1 - mx));
    out[(long long)n * 2 + 0] = o0 - ls;
    out[(long long)n * 2 + 1] = o1 - ls;
}

// ---------------------------------------------------------------------------
extern "C" void kernel_launch(void* const* d_in, const int* in_sizes, int n_in,
                              void* d_out, int out_size, void* d_ws, size_t ws_size,
                              hipStream_t stream) {
    const float* x     = (const float*)d_in[0];
    const int*   eidx  = (const int*)  d_in[1];   // [2, E]
    const int*   etype = (const int*)  d_in[2];   // [E]
    const float* W1    = (const float*)d_in[3];   // [16,7,16]
    const float* root1 = (const float*)d_in[4];   // [7,16]
    const float* b1    = (const float*)d_in[5];   // [16]
    const float* W2    = (const float*)d_in[6];   // [16,16,2]
    const float* root2 = (const float*)d_in[7];   // [16,2]
    const float* b2    = (const float*)d_in[8];   // [2]

    int nN = in_sizes[0] / 7;       // 200000
    int nE = in_sizes[2];           // 6400000
    const int* src = eidx;
    const int* dst = eidx + nE;

    // Workspace layout (floats): agg1[16N] | deg[N] | h[16N] | hW2[32N] | agg2[2N]
    float* ws   = (float*)d_ws;
    float* agg1 = ws;
    float* deg  = agg1 + (long long)nN * 16;
    float* h    = deg  + nN;
    float* hW2  = h    + (long long)nN * 16;
    float* agg2 = hW2  + (long long)nN * 32;
    float* out  = (float*)d_out;

    long long zn = 67LL * nN;  // zero entire workspace footprint
    zero_ws_kernel<<<(int)((zn + 1023) / 1024), 1024, 0, stream>>>(ws, zn);

    long long tasks = (long long)nE * 16;
    int gl1 = (int)((tasks + 256LL * L1_ITER - 1) / (256LL * L1_ITER));
    edge_l1_kernel<<<gl1, 256, 0, stream>>>(x, src, dst, etype, W1, agg1, deg, nE);

    rdeg_kernel<<<(nN + 255) / 256, 256, 0, stream>>>(deg, nN);

    int nTiles = (nN + 15) / 16;       // 12500 (N divisible by 16)
    int gT = (nTiles + 7) / 8;
    node_l1_kernel<<<gT, 256, 0, stream>>>(x, root1, b1, agg1, deg, h, nTiles);
    hw2_kernel<<<gT, 256, 0, stream>>>(h, W2, hW2, nTiles);

    edge_l2_kernel<<<(nE + 255) / 256, 256, 0, stream>>>(src, dst, etype, hW2, agg2, nE);
    final_kernel<<<(nN + 255) / 256, 256, 0, stream>>>(h, root2, b2, agg2, deg, out, nN);
}